// LocalCorr_59133109731737
// MI455X (gfx1250) — compile-verified
//
#include <hip/hip_runtime.h>
#include <math.h>

// ---- problem dims (match reference setup_inputs) ----
#define Bb 8
#define Cc 64
#define Hh 180
#define Ww 320
#define HW (Hh * Ww)

// ---- tiling ----
#define TILE_W   32          // pixels per block (2 waves x 16-pixel strips)
#define NWAVE    2
#define NTHREADS 64
#define PITCH    42          // LDS row pitch (floats): 2*PITCH%64=20 -> half-waves hit disjoint banks
#define DPITCH   18          // D-scratch row pitch (floats)

typedef __attribute__((ext_vector_type(2))) float v2f;
typedef __attribute__((ext_vector_type(8))) float v8f;

__device__ __forceinline__ v8f wmma4(v2f a, v2f b, v8f c) {
    // D(16x16,f32) += A(16x4,f32) * B(4x16,f32)
    return __builtin_amdgcn_wmma_f32_16x16x4_f32(false, a, false, b, (short)0, c,
                                                 false, false);
}

__device__ __forceinline__ int refl(int i, int n) {
    i = (i < 0) ? -i : i;
    i = (i >= n) ? (2 * n - 2 - i) : i;
    return i;
}

__global__ __launch_bounds__(NTHREADS) void localcorr_kernel(
    const float* __restrict__ nbr, const float* __restrict__ ref,
    float* __restrict__ out) {
    // raw data tiles (norms folded in later via weights)
    __shared__ float nbrT[3 * Cc * PITCH];          // [di][c][x], x = q+2, q in [-1,TILE_W]
    __shared__ float refT[Cc * PITCH];              // [c][q],   q in [0,TILE_W)
    __shared__ float Dsc[NWAVE * 3 * 16 * DPITCH];  // per-wave 16x16 corr matrices
    __shared__ float invN[3 * 48];                  // inv L2 norm of nbr pixels (idx q+2)
    __shared__ float invR[TILE_W];                  // inv L2 norm of ref pixels

    const int tid   = threadIdx.x;
    const int wbase = blockIdx.x * TILE_W;
    const int h     = blockIdx.y;
    const int b     = blockIdx.z;

    // ---------------- stage raw tiles (reflect padding) ----------------
    {
        const int NX = TILE_W + 2;  // 34 columns incl. halo
        for (int idx = tid; idx < 3 * Cc * NX; idx += NTHREADS) {
            int di = idx / (Cc * NX);
            int r  = idx - di * (Cc * NX);
            int c  = r / NX;
            int x  = r - c * NX;                       // 0..33  (q = x-1)
            int gy = refl(h - 1 + di, Hh);
            int gx = refl(wbase + x - 1, Ww);
            nbrT[(di * Cc + c) * PITCH + (x + 1)] =
                nbr[(b * Cc + c) * HW + gy * Ww + gx];
        }
        for (int idx = tid; idx < Cc * TILE_W; idx += NTHREADS) {
            int c = idx >> 5;
            int q = idx & 31;
            refT[c * PITCH + q] = ref[(b * Cc + c) * HW + h * Ww + (wbase + q)];
        }
    }
    __syncthreads();

    // ---------------- per-pixel inverse L2 norms over channels ----------------
    for (int p = tid; p < 3 * (TILE_W + 2); p += NTHREADS) {
        int di = p / (TILE_W + 2);
        int x  = p - di * (TILE_W + 2);
        const float* col = &nbrT[di * Cc * PITCH + (x + 1)];
        float s = 0.f;
#pragma unroll 8
        for (int c = 0; c < Cc; ++c) { float v = col[c * PITCH]; s += v * v; }
        invN[di * 48 + (x + 1)] = 1.f / fmaxf(sqrtf(s), 1e-12f);
    }
    for (int p = tid; p < TILE_W; p += NTHREADS) {
        float s = 0.f;
#pragma unroll 8
        for (int c = 0; c < Cc; ++c) { float v = refT[c * PITCH + p]; s += v * v; }
        invR[p] = 1.f / fmaxf(sqrtf(s), 1e-12f);
    }
    __syncthreads();

    const int l    = tid & 31;
    const int wave = tid >> 5;
    const int sb   = wave * 16;  // strip base pixel within tile
    const int m    = l & 15;     // this lane's pixel / matrix row-col id
    const int half = l >> 4;     // half-wave: K-offset +2 in WMMA operand layouts

    const v8f vzero = {0.f, 0.f, 0.f, 0.f, 0.f, 0.f, 0.f, 0.f};

    // ---------------- correlation: D[di] = Ref(16xK) * Nbr_di(Kx16), K=64 ----------------
    v8f accD[3];
    accD[0] = vzero; accD[1] = vzero; accD[2] = vzero;
#pragma unroll
    for (int kk = 0; kk < 16; ++kk) {
        int c0 = 4 * kk + 2 * half;
        v2f a;
        a.x = refT[c0 * PITCH + sb + m];
        a.y = refT[(c0 + 1) * PITCH + sb + m];
#pragma unroll
        for (int di = 0; di < 3; ++di) {
            const float* pb = &nbrT[(di * Cc + c0) * PITCH + (sb + m + 2)];
            v2f bm; bm.x = pb[0]; bm.y = pb[PITCH];
            accD[di] = wmma4(a, bm, accD[di]);
        }
    }
    // spill D to LDS for tri-diagonal extraction (same-wave DS ops are in-order)
    float* Dx = &Dsc[wave * 3 * 16 * DPITCH];
#pragma unroll
    for (int di = 0; di < 3; ++di)
#pragma unroll
        for (int j = 0; j < 8; ++j)
            Dx[(di * 16 + j + 8 * half) * DPITCH + m] = accD[di][j];

    // strip-edge raw dots (n = -1 for m=0, n = 16 for m=15) via VALU + butterfly
    float dotL[3], dotR[3];
    {
        int cA = 2 * l;  // 32 lanes x 2 channels = 64
        float rA0 = refT[cA * PITCH + sb];
        float rA1 = refT[(cA + 1) * PITCH + sb];
        float rB0 = refT[cA * PITCH + sb + 15];
        float rB1 = refT[(cA + 1) * PITCH + sb + 15];
#pragma unroll
        for (int di = 0; di < 3; ++di) {
            const float* pn = &nbrT[(di * Cc + cA) * PITCH];
            dotL[di] = rA0 * pn[sb + 1]  + rA1 * pn[PITCH + sb + 1];   // q = sb-1
            dotR[di] = rB0 * pn[sb + 18] + rB1 * pn[PITCH + sb + 18];  // q = sb+16
        }
#pragma unroll
        for (int off = 16; off > 0; off >>= 1) {
#pragma unroll
            for (int di = 0; di < 3; ++di) {
                dotL[di] += __shfl_xor(dotL[di], off, 32);
                dotR[di] += __shfl_xor(dotR[di], off, 32);
            }
        }
    }

    // ---------------- extract tri-diagonals, scale, softmax, fuse norms ----------------
    float sw[9];  // softmax weight * inv-norm of the matching neighbor
    {
        float d[9], inb[9];
        float ivr = invR[sb + m];
#pragma unroll
        for (int di = 0; di < 3; ++di) {
#pragma unroll
            for (int dj = -1; dj <= 1; ++dj) {
                int n  = m + dj;
                int nc = n < 0 ? 0 : (n > 15 ? 15 : n);
                float raw = Dx[(di * 16 + m) * DPITCH + nc];
                raw = (n < 0)  ? dotL[di] : raw;
                raw = (n > 15) ? dotR[di] : raw;
                float ib = invN[di * 48 + (sb + m + dj + 2)];
                int idx  = di * 3 + dj + 1;
                inb[idx] = ib;
                d[idx]   = raw * ivr * ib;
            }
        }
        float mx = d[0];
#pragma unroll
        for (int i = 1; i < 9; ++i) mx = fmaxf(mx, d[i]);
        float ssum = 0.f, e[9];
#pragma unroll
        for (int i = 0; i < 9; ++i) { e[i] = __expf(d[i] - mx); ssum += e[i]; }
        float rs = 1.f / ssum;
#pragma unroll
        for (int i = 0; i < 9; ++i) sw[i] = e[i] * rs * inb[i];
    }

    // ---------------- aggregation: Agg(16ch x 16px) = NbrRaw * S(tri-diag) ----------------
    const float eScale = ((m == 0) | (m == 15)) ? 1.f : 0.f;  // predicated edge fixup
    const int   xe     = (m == 15) ? (sb + 18) : (sb + 1);
    float eW[3];
#pragma unroll
    for (int di = 0; di < 3; ++di)
        eW[di] = ((m == 15) ? sw[di * 3 + 2] : sw[di * 3 + 0]) * eScale;

    const size_t outBase =
        (size_t)(b * Cc) * HW + (size_t)h * Ww + (size_t)(wbase + sb + m);

#pragma unroll
    for (int cc = 0; cc < 4; ++cc) {
        v8f acc = vzero;
#pragma unroll
        for (int di = 0; di < 3; ++di) {
            float sm1 = sw[di * 3 + 0], s0v = sw[di * 3 + 1], sp1 = sw[di * 3 + 2];
#pragma unroll
            for (int kk = 0; kk < 4; ++kk) {
                int kn0 = 4 * kk + 2 * half;
                // banded S column m: S[n][m] = sw_{n-m} for |n-m|<=1 else 0
                v2f bm;
                {
                    int dj0 = kn0 - m;
                    float v = 0.f;
                    v = (dj0 == -1) ? sm1 : v;
                    v = (dj0 ==  0) ? s0v : v;
                    v = (dj0 ==  1) ? sp1 : v;
                    bm.x = v;
                    int dj1 = dj0 + 1;
                    float u = 0.f;
                    u = (dj1 == -1) ? sm1 : u;
                    u = (dj1 ==  0) ? s0v : u;
                    u = (dj1 ==  1) ? sp1 : u;
                    bm.y = u;
                }
                const float* pa =
                    &nbrT[(di * Cc + 16 * cc + m) * PITCH + (sb + 2 + kn0)];
                v2f am; am.x = pa[0]; am.y = pa[1];
                acc = wmma4(am, bm, acc);
            }
        }
        // out-of-band neighbors of the strip edges (n = -1 / n = 16)
#pragma unroll
        for (int j = 0; j < 8; ++j) {
            float addv = 0.f;
#pragma unroll
            for (int di = 0; di < 3; ++di)
                addv += eW[di] * nbrT[(di * Cc + 16 * cc + j + 8 * half) * PITCH + xe];
            acc[j] += addv;
        }
        // store: vgpr j holds channel (16*cc + j + 8*half), pixel (sb+m)
#pragma unroll
        for (int j = 0; j < 8; ++j) {
            int ch = 16 * cc + j + 8 * half;
            out[outBase + (size_t)ch * HW] = acc[j];
        }
    }
}

extern "C" void kernel_launch(void* const* d_in, const int* in_sizes, int n_in,
                              void* d_out, int out_size, void* d_ws, size_t ws_size,
                              hipStream_t stream) {
    (void)in_sizes; (void)n_in; (void)d_ws; (void)ws_size; (void)out_size;
    const float* nbr = (const float*)d_in[0];
    const float* ref = (const float*)d_in[1];
    float* out = (float*)d_out;

    dim3 grid(Ww / TILE_W, Hh, Bb);
    dim3 block(NTHREADS);
    hipLaunchKernelGGL(localcorr_kernel, grid, block, 0, stream, nbr, ref, out);
}